// MultiheadAttentionLinear_46557445488850
// MI455X (gfx1250) — compile-verified
//
#include <hip/hip_runtime.h>
#include <hip/hip_bf16.h>

// ---------------------------------------------------------------------------
// Linear attention (Katharopoulos) for MI455X / gfx1250.
//   B=4, S=1024, M=512, H=8, D=64.  R = B*S = 4096 flattened rows.
// fp32 WMMA (V_WMMA_F32_16X16X4_F32) + Tensor Data Mover weight staging.
// ---------------------------------------------------------------------------

typedef __attribute__((ext_vector_type(2))) float v2f;
typedef __attribute__((ext_vector_type(8))) float v8f;
typedef unsigned int u32x4 __attribute__((ext_vector_type(4)));
typedef int          i32x8 __attribute__((ext_vector_type(8)));
typedef int          i32x4 __attribute__((ext_vector_type(4)));

#define RR        4096          // B*S
#define MM        512
#define DD        64
#define HH        8
#define SS        1024
#define KC        64            // K-chunk staged in LDS per TDM transfer
#define HEADBUF   (RR * DD)     // floats per per-head q/k/v slab

__device__ __forceinline__ float elu1(float x) {
    return (x > 0.0f) ? (x + 1.0f) : __expf(x);   // elu(x)+1
}

// Truncate a generic pointer to a 32-bit LDS byte offset (low 32 bits of the
// shared-aperture address are the LDS offset on gfx1250).
__device__ __forceinline__ unsigned lds_off32(const void* p) {
    return (unsigned)(unsigned long long)p;
}

// ---------------------------------------------------------------------------
// Issue a TDM 2D tile load: rows x cols fp32 elements, global row stride
// `row_stride` elements, into LDS at byte offset `lds_off`.
// D# packing per CDNA5 ISA ch. 8 (groups 0/1; groups 2/3 zero => 2D).
// 6-arg builtin flavor (clang-23 / therock-10.0 headers).
// ---------------------------------------------------------------------------
__device__ __forceinline__ void tdm_load_2d(unsigned lds_off, const void* gptr,
                                            unsigned rows, unsigned cols,
                                            unsigned row_stride) {
    unsigned long long ga = (unsigned long long)gptr;
    u32x4 g0;
    g0[0] = 1u;                                         // count=1, user D#
    g0[1] = lds_off;                                    // LDS byte address
    g0[2] = (unsigned)(ga & 0xFFFFFFFFu);               // global addr [31:0]
    g0[3] = (unsigned)((ga >> 32) & 0x01FFFFFFu)        // global addr [56:32]
          | (2u << 30);                                 // type = 2 ("image")
    i32x8 g1;
    g1[0] = (int)(2u << 16);                            // data_size = 4 bytes
    g1[1] = (int)((cols & 0xFFFFu) << 16);              // tensor_dim0 lo16
    g1[2] = (int)((cols >> 16) | ((rows & 0xFFFFu) << 16)); // dim0 hi / dim1 lo
    g1[3] = (int)((rows >> 16) | ((cols & 0xFFFFu) << 16)); // dim1 hi / tile_dim0
    g1[4] = (int)(rows & 0xFFFFu);                      // tile_dim1 (tile_dim2=0)
    g1[5] = (int)row_stride;                            // tensor_dim0_stride lo32
    g1[6] = 0;                                          // stride hi / dim1_stride lo
    g1[7] = 0;
    i32x4 g2 = {0, 0, 0, 0};
    i32x4 g3 = {0, 0, 0, 0};
    i32x8 g4 = {0, 0, 0, 0, 0, 0, 0, 0};
    __builtin_amdgcn_tensor_load_to_lds(g0, g1, g2, g3, g4, 0);
}

// ---------------------------------------------------------------------------
// Kernel A: fused Q/K/V projection + elu()+1 feature map (on q,k only).
//   X : [4096, 512]      W* : [8, 512, 64]      out : q/k/v as [h, r, 64]
// Block = 128 rows x 64 cols for one (weight, head) pair; 8 waves, each wave
// a 16x64 tile (4 accumulators). Weight panel TDM-staged into LDS, K-chunked
// and double buffered; A rows streamed from global (L2 resident).
// ---------------------------------------------------------------------------
__global__ __launch_bounds__(256) void qkv_proj_wmma(
    const float* __restrict__ X,
    const float* __restrict__ Wq,
    const float* __restrict__ Wk,
    const float* __restrict__ Wv,
    float* __restrict__ qf,
    float* __restrict__ kf,
    float* __restrict__ vf)
{
    __shared__ __align__(16) float ldsB[2][KC * 64];   // 2 x 16 KB

    const int wave = threadIdx.x >> 5;
    const int lane = threadIdx.x & 31;
    const int m    = lane & 15;        // A-row / B-col index within tile
    const int half = lane >> 4;        // K sub-pair selector

    const int cb = blockIdx.x % 24;    // (weight, head): w*8 + h
    const int rb = blockIdx.x / 24;    // 0..31  (128-row blocks)
    const int w  = cb >> 3;            // 0:q 1:k 2:v
    const int h  = cb & 7;

    const float* W   = (w == 0) ? Wq : (w == 1) ? Wk : Wv;
    const float* Wh  = W + (size_t)h * MM * DD;         // [512][64], row stride 64
    float*       out = (w == 0) ? qf : (w == 1) ? kf : vf;

    const int r0 = rb * 128 + wave * 16;
    const float* Arow = X + (size_t)(r0 + m) * MM;

    const unsigned lb0 = lds_off32(&ldsB[0][0]);
    const unsigned lb1 = lds_off32(&ldsB[1][0]);

    if (wave == 0) tdm_load_2d(lb0, Wh, KC, 64, 64);    // preload chunk 0

    v8f acc0 = {}, acc1 = {}, acc2 = {}, acc3 = {};

    const int NCH = MM / KC;                            // 8 chunks
    for (int c = 0; c < NCH; ++c) {
        if (wave == 0) {
            if (c + 1 < NCH) {
                tdm_load_2d((c + 1) & 1 ? lb1 : lb0,
                            Wh + (size_t)(c + 1) * KC * 64, KC, 64, 64);
                __builtin_amdgcn_s_wait_tensorcnt(1);   // chunk c landed
            } else {
                __builtin_amdgcn_s_wait_tensorcnt(0);
            }
        }
        __syncthreads();

        const float* Bc = &ldsB[c & 1][0];
        const float* Ac = Arow + c * KC;
#pragma unroll
        for (int k = 0; k < KC; k += 4) {
            const int kb = k + 2 * half;
            v2f a = *(const v2f*)(Ac + kb);
            v2f b0, b1, b2, b3;
            b0.x = Bc[kb * 64 +  0 + m];  b0.y = Bc[(kb + 1) * 64 +  0 + m];
            b1.x = Bc[kb * 64 + 16 + m];  b1.y = Bc[(kb + 1) * 64 + 16 + m];
            b2.x = Bc[kb * 64 + 32 + m];  b2.y = Bc[(kb + 1) * 64 + 32 + m];
            b3.x = Bc[kb * 64 + 48 + m];  b3.y = Bc[(kb + 1) * 64 + 48 + m];
            acc0 = __builtin_amdgcn_wmma_f32_16x16x4_f32(false, a, false, b0, (short)0, acc0, false, false);
            acc1 = __builtin_amdgcn_wmma_f32_16x16x4_f32(false, a, false, b1, (short)0, acc1, false, false);
            acc2 = __builtin_amdgcn_wmma_f32_16x16x4_f32(false, a, false, b2, (short)0, acc2, false, false);
            acc3 = __builtin_amdgcn_wmma_f32_16x16x4_f32(false, a, false, b3, (short)0, acc3, false, false);
        }
        __syncthreads();
    }

    // Epilogue: C/D VGPR j holds (row = r0 + j + 8*half, col-subtile n0 + m)
    const size_t obase = (size_t)h * HEADBUF + m;
#pragma unroll
    for (int j = 0; j < 8; ++j) {
        const int row = r0 + j + 8 * half;
        float x0 = acc0[j], x1 = acc1[j], x2 = acc2[j], x3 = acc3[j];
        if (w < 2) { x0 = elu1(x0); x1 = elu1(x1); x2 = elu1(x2); x3 = elu1(x3); }
        float* orow = out + obase + (size_t)row * DD;
        orow[ 0] = x0;  orow[16] = x1;  orow[32] = x2;  orow[48] = x3;
    }
}

// ---------------------------------------------------------------------------
// Kernel B: causal cumulative-state scan per (h, b).
//   Thread j owns state column S[:,j] in 64 VGPRs.
//   Per step t:  S[:,j] += k_t * v_t[j];  num_j = q_t . S[:,j];
//                zc += k_t;               Z = q_t . zc   (fused FMA loop)
// Writes normalized heads straight into combined [b, s, h*64+d] layout.
// ---------------------------------------------------------------------------
__global__ __launch_bounds__(64) void linattn_scan(
    const float* __restrict__ qf,
    const float* __restrict__ kf,
    const float* __restrict__ vf,
    float* __restrict__ comb)
{
    const int hb = blockIdx.x;       // 0..31
    const int h  = hb >> 2;
    const int b  = hb & 3;
    const int j  = threadIdx.x;      // 0..63

    __shared__ float sh_q[DD];
    __shared__ float sh_k[DD];
    __shared__ float sh_zc[DD];

    float Scol[DD];
#pragma unroll
    for (int i = 0; i < DD; ++i) Scol[i] = 0.0f;
    sh_zc[j] = 0.0f;
    __syncthreads();

    const size_t base  = (size_t)h * HEADBUF + (size_t)b * SS * DD;
    const size_t cbase = (size_t)b * SS * MM + (size_t)h * DD + j;

    for (int t = 0; t < SS; ++t) {
        const size_t off = base + (size_t)t * DD + j;
        const float qj = qf[off];
        const float kj = kf[off];
        const float vj = vf[off];
        sh_q[j]  = qj;
        sh_k[j]  = kj;
        sh_zc[j] += kj;
        __syncthreads();

        float num = 0.0f, Z = 0.0f;
#pragma unroll
        for (int i = 0; i < DD; ++i) {
            Scol[i] = fmaf(sh_k[i], vj, Scol[i]);
            num     = fmaf(sh_q[i], Scol[i], num);
            Z       = fmaf(sh_q[i], sh_zc[i], Z);
        }
        comb[cbase + (size_t)t * MM] = num / Z;
        __syncthreads();   // protect sh_* from next iteration's writes
    }
}

// ---------------------------------------------------------------------------
// Kernel C: output projection  out[4096,512] = comb[4096,512] @ Wo[512,512]
// Same blocking as kernel A; Wo stripe (512 x 64, row stride 512) TDM-staged.
// ---------------------------------------------------------------------------
__global__ __launch_bounds__(256) void out_proj_wmma(
    const float* __restrict__ comb,
    const float* __restrict__ Wo,
    float* __restrict__ out)
{
    __shared__ __align__(16) float ldsB[2][KC * 64];

    const int wave = threadIdx.x >> 5;
    const int lane = threadIdx.x & 31;
    const int m    = lane & 15;
    const int half = lane >> 4;

    const int cb = blockIdx.x & 7;     // 8 column blocks of 64
    const int rb = blockIdx.x >> 3;    // 32 row blocks of 128
    const int c0 = cb * 64;

    const float* Wc = Wo + c0;         // element (k, n) at Wc[k*512 + n]
    const int r0 = rb * 128 + wave * 16;
    const float* Arow = comb + (size_t)(r0 + m) * MM;

    const unsigned lb0 = lds_off32(&ldsB[0][0]);
    const unsigned lb1 = lds_off32(&ldsB[1][0]);

    if (wave == 0) tdm_load_2d(lb0, Wc, KC, 64, MM);

    v8f acc0 = {}, acc1 = {}, acc2 = {}, acc3 = {};

    const int NCH = MM / KC;
    for (int c = 0; c < NCH; ++c) {
        if (wave == 0) {
            if (c + 1 < NCH) {
                tdm_load_2d((c + 1) & 1 ? lb1 : lb0,
                            Wc + (size_t)(c + 1) * KC * MM, KC, 64, MM);
                __builtin_amdgcn_s_wait_tensorcnt(1);
            } else {
                __builtin_amdgcn_s_wait_tensorcnt(0);
            }
        }
        __syncthreads();

        const float* Bc = &ldsB[c & 1][0];
        const float* Ac = Arow + c * KC;
#pragma unroll
        for (int k = 0; k < KC; k += 4) {
            const int kb = k + 2 * half;
            v2f a = *(const v2f*)(Ac + kb);
            v2f b0, b1, b2, b3;
            b0.x = Bc[kb * 64 +  0 + m];  b0.y = Bc[(kb + 1) * 64 +  0 + m];
            b1.x = Bc[kb * 64 + 16 + m];  b1.y = Bc[(kb + 1) * 64 + 16 + m];
            b2.x = Bc[kb * 64 + 32 + m];  b2.y = Bc[(kb + 1) * 64 + 32 + m];
            b3.x = Bc[kb * 64 + 48 + m];  b3.y = Bc[(kb + 1) * 64 + 48 + m];
            acc0 = __builtin_amdgcn_wmma_f32_16x16x4_f32(false, a, false, b0, (short)0, acc0, false, false);
            acc1 = __builtin_amdgcn_wmma_f32_16x16x4_f32(false, a, false, b1, (short)0, acc1, false, false);
            acc2 = __builtin_amdgcn_wmma_f32_16x16x4_f32(false, a, false, b2, (short)0, acc2, false, false);
            acc3 = __builtin_amdgcn_wmma_f32_16x16x4_f32(false, a, false, b3, (short)0, acc3, false, false);
        }
        __syncthreads();
    }

#pragma unroll
    for (int j = 0; j < 8; ++j) {
        const int row = r0 + j + 8 * half;
        float* orow = out + (size_t)row * MM + c0 + m;
        orow[ 0] = acc0[j];
        orow[16] = acc1[j];
        orow[32] = acc2[j];
        orow[48] = acc3[j];
    }
}

// ---------------------------------------------------------------------------
// Inputs (setup_inputs order): X, mask(unused), Wq, Wk, Wv, Wo.
// Output fp32 [4,1024,512]. Workspace: q | k | v | combined, 8 MB each.
// ---------------------------------------------------------------------------
extern "C" void kernel_launch(void* const* d_in, const int* in_sizes, int n_in,
                              void* d_out, int out_size, void* d_ws, size_t ws_size,
                              hipStream_t stream) {
    const float* X  = (const float*)d_in[0];
    // d_in[1] = mask, unused by the reference module
    const float* Wq = (const float*)d_in[2];
    const float* Wk = (const float*)d_in[3];
    const float* Wv = (const float*)d_in[4];
    const float* Wo = (const float*)d_in[5];
    float* out = (float*)d_out;

    float* ws = (float*)d_ws;
    const size_t BUF = (size_t)RR * MM;   // 2,097,152 floats = 8 MB
    float* qf   = ws + 0 * BUF;
    float* kf   = ws + 1 * BUF;
    float* vf   = ws + 2 * BUF;
    float* comb = ws + 3 * BUF;

    qkv_proj_wmma<<<dim3(24 * 32), dim3(256), 0, stream>>>(X, Wq, Wk, Wv, qf, kf, vf);
    linattn_scan<<<dim3(32), dim3(64), 0, stream>>>(qf, kf, vf, comb);
    out_proj_wmma<<<dim3(8 * 32), dim3(256), 0, stream>>>(comb, Wo, out);
}